// MyModel_78950088835490
// MI455X (gfx1250) — compile-verified
//
#include <hip/hip_runtime.h>
#include <cmath>

// CDNA5 / gfx1250, wave32. f32 WMMA: D(16x16) = A(16x4) * B(4x16) + C.
typedef __attribute__((ext_vector_type(2))) float v2f;
typedef __attribute__((ext_vector_type(4))) float v4f;
typedef __attribute__((ext_vector_type(8))) float v8f;

#define NB   2048      // batch
#define WD   128       // hidden width
#define DIN  131       // 1 + 3 + 127
#define LSX  132       // padded LDS row stride (multiple of 4, shifts banks)
#define GRP  16        // examples per workgroup (M tile)
#define NTH  256       // 8 waves

// flat float output offsets, reference tuple order
#define OFF_RES   0
#define OFF_FMEAN (OFF_RES   + NB)
#define OFF_FLOG  (OFF_FMEAN + NB*WD)
#define OFF_COEF  (OFF_FLOG  + NB*WD)
#define OFF_D1C   (OFF_COEF  + NB)
#define OFF_D2C   (OFF_D1C   + NB)
#define OFF_D1O   (OFF_D2C   + NB)
#define OFF_D2O   (OFF_D1O   + NB*3)
#define OFF_D1F   (OFF_D2O   + NB*9)
#define OFF_D2F   (OFF_D1F   + NB*WD)

__device__ __forceinline__ v8f wmma4(v2f a, v2f b, v8f c) {
  // (neg_a, A, neg_b, B, c_mod, C, reuse_a, reuse_b)
  return __builtin_amdgcn_wmma_f32_16x16x4_f32(false, a, false, b, (short)0, c,
                                               false, false);
}

// ---------------- Fill kernel: streaming zeros for d2o and d2f ----------------
// d2o ≡ 0 (3x3 per example); d2f ≡ 0 except [0,0], written later by compute
// kernel (stream-ordered after this one). 16B nontemporal stores.
__global__ __launch_bounds__(NTH) void zero_fill_kernel(float* __restrict__ out) {
  const size_t gsz = (size_t)gridDim.x * NTH;
  size_t gtid = (size_t)blockIdx.x * NTH + threadIdx.x;

  v4f* d2o = (v4f*)(out + OFF_D2O);               // 18432 floats, 16B aligned
  const size_t n_d2o = (size_t)NB * 9 / 4;
  const v4f z = {0.f, 0.f, 0.f, 0.f};
  for (size_t i = gtid; i < n_d2o; i += gsz)
    __builtin_nontemporal_store(z, d2o + i);

  v4f* d2f = (v4f*)(out + OFF_D2F);               // 33.55M floats
  const size_t n_d2f = (size_t)NB * WD * WD / 4;
  for (size_t i = gtid; i < n_d2f; i += gsz)
    __builtin_nontemporal_store(z, d2f + i);
}

// Forward layer tile: D[e][n] = relu( sum_k A[e][k] * W[k*WD+n] + bias[n] )
// A in LDS (stride LSX). Wave 'wv' owns columns [wv*16, wv*16+16).
// Main loop is unguarded over 128 K; optional guarded tail covers k=128..130.
__device__ __forceinline__ void fwd_layer(const float* As, const float* __restrict__ Wg,
                                          const float* __restrict__ bg, float* Ds,
                                          bool tail131, int lane, int wv) {
  const int m    = lane & 15;          // A row (example)
  const int koff = (lane >> 4) << 1;   // 0 or 2
  const int n    = wv * 16 + (lane & 15);
  v8f acc = {0.f,0.f,0.f,0.f,0.f,0.f,0.f,0.f};
  for (int k0 = 0; k0 < WD; k0 += 4) {
    const int kb = k0 + koff;
    v2f a = *(const v2f*)(As + m * LSX + kb);
    v2f b;
    b.x = Wg[kb * WD + n];
    b.y = Wg[(kb + 1) * WD + n];
    acc = wmma4(a, b, acc);
  }
  if (tail131) {                       // k = 128..131 (x[131]==0 pad in A)
    const int kb = 128 + koff;
    v2f a = *(const v2f*)(As + m * LSX + kb);
    v2f b;
    b.x = (kb     < DIN) ? Wg[kb * WD + n]       : 0.f;
    b.y = (kb + 1 < DIN) ? Wg[(kb + 1) * WD + n] : 0.f;
    acc = wmma4(a, b, acc);
  }
  const float bias = bg[n];
  const int mbase = (lane >> 4) << 3;  // rows 0..7 or 8..15
  #pragma unroll
  for (int r = 0; r < 8; ++r) {
    float z = acc[r] + bias;
    Ds[(mbase + r) * LSX + n] = z > 0.f ? z : 0.f;
  }
}

// Backward layer tile: D[e][m'] = ( sum_n A[e][n] * W[m'*WD+n] ) * (H[e][m']>0)
__device__ __forceinline__ void bwd_layer(const float* As, const float* __restrict__ Wg,
                                          const float* Hs, float* Ds, int lane, int wv) {
  const int m    = lane & 15;
  const int koff = (lane >> 4) << 1;
  const int n    = wv * 16 + (lane & 15);   // output column m'
  v8f acc = {0.f,0.f,0.f,0.f,0.f,0.f,0.f,0.f};
  for (int k0 = 0; k0 < WD; k0 += 4) {
    const int kb = k0 + koff;
    v2f a = *(const v2f*)(As + m * LSX + kb);
    v2f b = *(const v2f*)(Wg + n * WD + kb);   // W^T fragment, contiguous b64
    acc = wmma4(a, b, acc);
  }
  const int mbase = (lane >> 4) << 3;
  #pragma unroll
  for (int r = 0; r < 8; ++r) {
    const int row = (mbase + r) * LSX + n;
    Ds[row] = (Hs[row] > 0.f) ? acc[r] : 0.f;
  }
}

// Input-gradient tile: gx[e][p] = sum_j A[e][j] * Wi[p*WD+j], p in [0,131)
__device__ __forceinline__ void gx_layer(const float* As, const float* __restrict__ Wig,
                                         float* Ds, int lane, int wv) {
  const int m    = lane & 15;
  const int koff = (lane >> 4) << 1;
  for (int nt = wv; nt * 16 < DIN; nt += 8) {   // 9 tiles; wave0 does nt=0 and 8
    const int p = nt * 16 + (lane & 15);
    v8f acc = {0.f,0.f,0.f,0.f,0.f,0.f,0.f,0.f};
    for (int k0 = 0; k0 < WD; k0 += 4) {
      const int kb = k0 + koff;
      v2f a = *(const v2f*)(As + m * LSX + kb);
      v2f b;
      if (p < DIN) b = *(const v2f*)(Wig + p * WD + kb);
      else         { b.x = 0.f; b.y = 0.f; }
      acc = wmma4(a, b, acc);
    }
    const int mbase = (lane >> 4) << 3;
    if (p < DIN) {
      #pragma unroll
      for (int r = 0; r < 8; ++r) Ds[(mbase + r) * LSX + p] = acc[r];
    }
  }
}

__global__ __launch_bounds__(NTH) void mlp_hess_kernel(
    const float* __restrict__ centers, const int* __restrict__ indices,
    const float* __restrict__ xs, const float* __restrict__ eps,
    const float* __restrict__ dict_kernel, const float* __restrict__ Wi,
    const float* __restrict__ bi, const float* __restrict__ Wh,
    const float* __restrict__ bh, const float* __restrict__ Wf,
    const float* __restrict__ bf, float* __restrict__ out) {
  __shared__ float xbuf[GRP * LSX];   // x on the way in, gx on the way out
  __shared__ float h1[GRP * LSX];
  __shared__ float h2[GRP * LSX];
  __shared__ float h3[GRP * LSX];
  __shared__ float h4[GRP * LSX];
  __shared__ float ft0s[GRP], cs[GRP], yv[GRP];

  const int tid  = threadIdx.x;
  const int lane = tid & 31;
  const int wv   = tid >> 5;
  const int e0   = blockIdx.x * GRP;

  // ---- Phase 0: gather, reparameterize, build x ----
  if (tid < GRP) {
    const int i = e0 + tid;
    cs[tid] = centers[i * 4 + 0];
    xbuf[tid * LSX + 1] = centers[i * 4 + 1];
    xbuf[tid * LSX + 2] = centers[i * 4 + 2];
    xbuf[tid * LSX + 3] = centers[i * 4 + 3];
    xbuf[tid * LSX + 131] = 0.f;   // K-pad
  }
  for (int idx = tid; idx < GRP * WD; idx += NTH) {
    const int e = idx >> 7, j = idx & 127;
    const int i = e0 + e;
    const int k = indices[i];
    const float mean = dict_kernel[k * 256 + j];
    const float lsg  = dict_kernel[k * 256 + 128 + j];
    const float ft   = mean + __expf(0.5f * lsg) * eps[k * WD + j];
    out[OFF_FMEAN + (size_t)i * WD + j] = mean;
    out[OFF_FLOG  + (size_t)i * WD + j] = lsg;
    if (j == 0) ft0s[e] = ft;
    else        xbuf[e * LSX + 3 + j] = ft;   // x[4..130] = ft[1..127]
  }
  __syncthreads();
  if (tid < GRP) {
    const float s = 1e-10f + __expf(-ft0s[tid]);
    xbuf[tid * LSX + 0] = cs[tid] * s;
  }
  __syncthreads();

  // ---- Forward MLP (WMMA f32 16x16x4) ----
  fwd_layer(xbuf, Wi, bi, h1, true, lane, wv);
  __syncthreads();
  fwd_layer(h1, Wh + 0 * WD * WD, bh + 0 * WD, h2, false, lane, wv);
  __syncthreads();
  fwd_layer(h2, Wh + 1 * WD * WD, bh + 1 * WD, h3, false, lane, wv);
  __syncthreads();
  fwd_layer(h3, Wh + 2 * WD * WD, bh + 2 * WD, h4, false, lane, wv);
  __syncthreads();

  // final dot: y[e] = h4[e]·Wf + bf
  if (tid < GRP) {
    float acc = bf[0];
    for (int j = 0; j < WD; ++j) acc += h4[tid * LSX + j] * Wf[j];
    yv[tid] = acc;
  }
  __syncthreads();

  // ---- Backward: d4 = Wf ⊙ (h4>0), in place ----
  for (int idx = tid; idx < GRP * WD; idx += NTH) {
    const int e = idx >> 7, j = idx & 127;
    const float h = h4[e * LSX + j];
    h4[e * LSX + j] = (h > 0.f) ? Wf[j] : 0.f;
  }
  __syncthreads();
  bwd_layer(h4, Wh + 2 * WD * WD, h3, h3, lane, wv);   // d3 over h3
  __syncthreads();
  bwd_layer(h3, Wh + 1 * WD * WD, h2, h2, lane, wv);   // d2 over h2
  __syncthreads();
  bwd_layer(h2, Wh + 0 * WD * WD, h1, h1, lane, wv);   // d1 over h1
  __syncthreads();
  gx_layer(h1, Wi, xbuf, lane, wv);                    // gx over xbuf
  __syncthreads();

  // ---- Finalize per example ----
  if (tid < GRP) {
    const int i = e0 + tid;
    const float g0  = xbuf[tid * LSX + 0];
    const float ex  = __expf(-ft0s[tid]);
    const float s   = 1e-10f + ex;
    const float c   = cs[tid];
    const float d1c = g0 * s;
    const float val = yv[tid];
    const float var = xs[i] - c;
    out[OFF_RES  + i] = val + d1c * var;   // d2c == 0 exactly (relu'' == 0)
    out[OFF_COEF + i] = val;
    out[OFF_D1C  + i] = d1c;
    out[OFF_D2C  + i] = 0.f;
    out[OFF_D1O + i * 3 + 0] = xbuf[tid * LSX + 1];
    out[OFF_D1O + i * 3 + 1] = xbuf[tid * LSX + 2];
    out[OFF_D1O + i * 3 + 2] = xbuf[tid * LSX + 3];
    const float d1f0 = -c * ex * g0;
    out[OFF_D1F + (size_t)i * WD + 0] = d1f0;
    // only nonzero Hessian entry; d2f zeroed by zero_fill_kernel (stream order)
    out[OFF_D2F + (size_t)i * WD * WD] = c * ex * g0;
  }
  for (int idx = tid; idx < GRP * WD; idx += NTH) {
    const int e = idx >> 7, j = idx & 127;
    if (j >= 1)
      out[OFF_D1F + (size_t)(e0 + e) * WD + j] = xbuf[e * LSX + 3 + j];
  }
}

extern "C" void kernel_launch(void* const* d_in, const int* in_sizes, int n_in,
                              void* d_out, int out_size, void* d_ws, size_t ws_size,
                              hipStream_t stream) {
  const float* centers     = (const float*)d_in[0];
  const int*   indices     = (const int*)  d_in[1];
  const float* xs          = (const float*)d_in[2];
  const float* eps         = (const float*)d_in[3];
  const float* dict_kernel = (const float*)d_in[4];
  const float* Wi          = (const float*)d_in[5];
  const float* bi          = (const float*)d_in[6];
  const float* Wh          = (const float*)d_in[7];
  const float* bh          = (const float*)d_in[8];
  const float* Wf          = (const float*)d_in[9];
  const float* bf          = (const float*)d_in[10];
  float* out = (float*)d_out;

  // 1) stream zeros for d2o/d2f at full-device occupancy (1024 WGs x 8 waves)
  hipLaunchKernelGGL(zero_fill_kernel, dim3(1024), dim3(NTH), 0, stream, out);
  // 2) compute; writes d2f[i][0][0] after the fill (stream-ordered)
  hipLaunchKernelGGL(mlp_hess_kernel, dim3(NB / GRP), dim3(NTH), 0, stream,
                     centers, indices, xs, eps, dict_kernel, Wi, bi, Wh, bh,
                     Wf, bf, out);
}